// DistortionLossDisparity_30434138259689
// MI455X (gfx1250) — compile-verified
//
#include <hip/hip_runtime.h>
#include <hip/hip_bf16.h>

typedef __attribute__((ext_vector_type(2))) float v2f;
typedef __attribute__((ext_vector_type(8))) float v8f;

#define NROWS 8192
#define DDIM  128
#define TEMP_INV 10.0f
#define CS 8                          // column stripes (grid.y)
#define WAVES_PER_WG 8
#define ROWS_PER_WAVE 32              // two 16-row A panels per wave
#define ROWS_PER_WG (WAVES_PER_WG * ROWS_PER_WAVE)   // 256

__global__ void k_zero(float* out) {
    if (threadIdx.x == 0) out[0] = 0.0f;
}

// One wave (32 lanes) per row: s[i] = q[i].q[j], labelterm[i] = |s - t[i].t[m(i,label)]| / TEMP
__global__ void k_precomp(const float* __restrict__ q, const float* __restrict__ t,
                          const int* __restrict__ labels, const int* __restrict__ jidx,
                          float* __restrict__ s_out, float* __restrict__ lbl_out) {
    const int i    = (blockIdx.x * blockDim.x + threadIdx.x) >> 5;
    const int lane = threadIdx.x & 31;
    if (i >= NROWS) return;

    const int j = jidx[i];
    const int k = labels[i];
    int mm;
    if (k == i)      mm = j;
    else if (j > i)  mm = (k > i && k <= j) ? k - 1 : k;
    else             mm = (k >= j && k < i) ? k + 1 : k;

    const float4* qi = (const float4*)(q + (size_t)i  * DDIM);
    const float4* qj = (const float4*)(q + (size_t)j  * DDIM);
    const float4* ti = (const float4*)(t + (size_t)i  * DDIM);
    const float4* tm = (const float4*)(t + (size_t)mm * DDIM);

    float4 a = qi[lane], b = qj[lane];
    float s = a.x * b.x + a.y * b.y + a.z * b.z + a.w * b.w;
    float4 c = ti[lane], d = tm[lane];
    float g = c.x * d.x + c.y * d.y + c.z * d.z + c.w * d.w;

    #pragma unroll
    for (int off = 16; off >= 1; off >>= 1) {
        s += __shfl_xor(s, off, 32);
        g += __shfl_xor(g, off, 32);
    }
    if (lane == 0) {
        s_out[i]   = s;
        lbl_out[i] = fabsf(s - g) * TEMP_INV;
    }
}

// Main: streaming T = t*t^T via V_WMMA_F32_16X16X4_F32 with online logsumexp of |s - T|/TEMP.
// Each wave holds TWO resident 16x128 A panels (32 rows) so every B fragment feeds 2 WMMAs:
// halves WGP-cache bandwidth per FLOP and gives two independent accumulator chains.
__global__ __launch_bounds__(256, 1)
void k_main(const float* __restrict__ t, const float* __restrict__ s_in,
            float* __restrict__ pmax, float* __restrict__ psum) {
    const int lane = threadIdx.x & 31;
    const int lo   = lane & 15;
    const int hi   = lane >> 4;
    const int wave = threadIdx.x >> 5;
    const int rowBase  = (blockIdx.x * WAVES_PER_WG + wave) * ROWS_PER_WAVE;
    const int stripe   = blockIdx.y;
    const int colStart = stripe * (NROWS / CS);
    const int colSteps = (NROWS / CS) / 16;     // 64

    // Resident A fragments (ISA 7.12.2 A 16x4 layout): lanes 0-15 K={4kk,4kk+1}, lanes 16-31 K={4kk+2,4kk+3}.
    const float* arow0 = t + (size_t)(rowBase + lo) * DDIM + 2 * hi;
    const float* arow1 = t + (size_t)(rowBase + 16 + lo) * DDIM + 2 * hi;
    v2f afrag0[32], afrag1[32];
    #pragma unroll
    for (int kk = 0; kk < 32; ++kk) {
        afrag0[kk] = *(const v2f*)(arow0 + 4 * kk);
        afrag1[kk] = *(const v2f*)(arow1 + 4 * kk);
    }

    // s for the C-slot rows this lane covers (tile p, slot r -> row rowBase + 16*p + r + 8*hi)
    float sreg0[8], sreg1[8];
    #pragma unroll
    for (int r = 0; r < 8; ++r) {
        sreg0[r] = s_in[rowBase + r + 8 * hi];
        sreg1[r] = s_in[rowBase + 16 + r + 8 * hi];
    }

    float mx0[8], sm0[8], mx1[8], sm1[8];
    #pragma unroll
    for (int r = 0; r < 8; ++r) {
        mx0[r] = -__builtin_inff(); sm0[r] = 0.0f;
        mx1[r] = -__builtin_inff(); sm1[r] = 0.0f;
    }

    // B fragment base for this lane: tile cs, step kk lives at bbase + cs*16*DDIM + 4*kk
    const float* bbase = t + (size_t)(colStart + lo) * DDIM + 2 * hi;

    #pragma unroll 1
    for (int cstep = 0; cstep < colSteps; ++cstep) {
        const float* p = bbase + (size_t)cstep * 16 * DDIM;
        v8f c0 = {}, c1 = {};
        #pragma unroll
        for (int kk = 0; kk < 32; ++kk) {
            v2f b = *(const v2f*)(p + 4 * kk);
            c0 = __builtin_amdgcn_wmma_f32_16x16x4_f32(
                     false, afrag0[kk], false, b, (short)0, c0, false, false);
            c1 = __builtin_amdgcn_wmma_f32_16x16x4_f32(
                     false, afrag1[kk], false, b, (short)0, c1, false, false);
        }
        // online logsumexp update: one element per slot (col = tile col lo)
        #pragma unroll
        for (int r = 0; r < 8; ++r) {
            float x0  = fabsf(sreg0[r] - c0[r]) * TEMP_INV;
            float nm0 = fmaxf(mx0[r], x0);
            sm0[r] = sm0[r] * __expf(mx0[r] - nm0) + __expf(x0 - nm0);
            mx0[r] = nm0;

            float x1  = fabsf(sreg1[r] - c1[r]) * TEMP_INV;
            float nm1 = fmaxf(mx1[r], x1);
            sm1[r] = sm1[r] * __expf(mx1[r] - nm1) + __expf(x1 - nm1);
            mx1[r] = nm1;
        }
    }

    // Merge (max,sum) across the 16 lanes that share each row.
    #pragma unroll
    for (int r = 0; r < 8; ++r) {
        #pragma unroll
        for (int off = 8; off >= 1; off >>= 1) {
            float om0 = __shfl_xor(mx0[r], off, 16);
            float os0 = __shfl_xor(sm0[r], off, 16);
            float nm0 = fmaxf(mx0[r], om0);
            sm0[r] = sm0[r] * __expf(mx0[r] - nm0) + os0 * __expf(om0 - nm0);
            mx0[r] = nm0;

            float om1 = __shfl_xor(mx1[r], off, 16);
            float os1 = __shfl_xor(sm1[r], off, 16);
            float nm1 = fmaxf(mx1[r], om1);
            sm1[r] = sm1[r] * __expf(mx1[r] - nm1) + os1 * __expf(om1 - nm1);
            mx1[r] = nm1;
        }
    }
    if (lo == 0) {
        #pragma unroll
        for (int r = 0; r < 8; ++r) {
            const int row0 = rowBase + r + 8 * hi;
            const int row1 = rowBase + 16 + r + 8 * hi;
            pmax[(size_t)stripe * NROWS + row0] = mx0[r];
            psum[(size_t)stripe * NROWS + row0] = sm0[r];
            pmax[(size_t)stripe * NROWS + row1] = mx1[r];
            psum[(size_t)stripe * NROWS + row1] = sm1[r];
        }
    }
}

__global__ void k_finalize(const float* __restrict__ pmax, const float* __restrict__ psum,
                           const float* __restrict__ lbl, float* __restrict__ out) {
    const int i = blockIdx.x * blockDim.x + threadIdx.x;
    float term = 0.0f;
    if (i < NROWS) {
        float M = -__builtin_inff();
        #pragma unroll
        for (int st = 0; st < CS; ++st) M = fmaxf(M, pmax[st * NROWS + i]);
        float S = 0.0f;
        #pragma unroll
        for (int st = 0; st < CS; ++st) S += psum[st * NROWS + i] * __expf(pmax[st * NROWS + i] - M);
        const float lse = M + __logf(S);
        term = (lse - lbl[i]) * (1.0f / (float)NROWS);
    }
    #pragma unroll
    for (int off = 16; off >= 1; off >>= 1) term += __shfl_xor(term, off, 32);
    if ((threadIdx.x & 31) == 0) atomicAdd(out, term);
}

extern "C" void kernel_launch(void* const* d_in, const int* in_sizes, int n_in,
                              void* d_out, int out_size, void* d_ws, size_t ws_size,
                              hipStream_t stream) {
    (void)in_sizes; (void)n_in; (void)out_size; (void)ws_size;
    const float* q      = (const float*)d_in[0];
    const float* t      = (const float*)d_in[1];
    const int*   labels = (const int*)d_in[2];
    const int*   jidx   = (const int*)d_in[3];
    float* out = (float*)d_out;

    float* ws     = (float*)d_ws;
    float* s_ws   = ws;                       // NROWS
    float* lbl_ws = ws + NROWS;               // NROWS
    float* pmax   = ws + 2 * NROWS;           // CS * NROWS
    float* psum   = ws + 2 * NROWS + (size_t)CS * NROWS;

    k_zero<<<1, 32, 0, stream>>>(out);
    k_precomp<<<NROWS / 8, 256, 0, stream>>>(q, t, labels, jidx, s_ws, lbl_ws);
    dim3 grid(NROWS / ROWS_PER_WG, CS);       // 32 x 8 = 256 workgroups
    k_main<<<grid, 256, 0, stream>>>(t, s_ws, pmax, psum);
    k_finalize<<<NROWS / 256, 256, 0, stream>>>(pmax, psum, lbl_ws, out);
}